// LIFNeuron_86122684220101
// MI455X (gfx1250) — compile-verified
//
#include <hip/hip_runtime.h>
#include <stdint.h>

// ---------------------------------------------------------------------------
// LIF neuron scan, (B,S,D) = (16,2048,1024) fp32.
// Parallelism = B*D = 16384 independent time-recurrences -> 512 wave32s.
// Memory-bound (256 MB traffic, ~1.5 flop/byte). Strategy: per-wave
// TDM (tensor_load_to_lds) triple-buffered pipeline staging 64-timestep
// x-tiles into LDS (TENSORcnt), consumed via ds loads; spikes streamed
// out with non-temporal stores so x stays resident in the 192 MB L2.
// ---------------------------------------------------------------------------

#define B_   16
#define S_   2048
#define D_   1024
#define TPB  128               // threads per block
#define DCH  (D_ / TPB)        // 8 d-chunks per batch  -> 128 blocks
#define NW   (TPB / 32)        // 4 waves per block
#define TT   64                // timesteps per TDM tile
#define NCH  (S_ / TT)         // 32 chunks
#define NBUF 3                 // TDM pipeline depth (LDS: 3*32KB = 96KB/block)

typedef unsigned int u32x4 __attribute__((ext_vector_type(4)));
typedef int          i32x8 __attribute__((ext_vector_type(8)));
typedef int          i32x4 __attribute__((ext_vector_type(4)));

typedef __attribute__((address_space(3))) float lds_f;

#if __has_builtin(__builtin_amdgcn_tensor_load_to_lds)
#define HAVE_TDM 1
#else
#define HAVE_TDM 0
#endif

// All waits carry a "memory" clobber: they both order the hardware counters
// and act as compiler barriers, so LDS reads between them can neither be
// hoisted above tile arrival nor considered reads of never-written memory.
__device__ __forceinline__ void wait_tensorcnt2() {
    asm volatile("s_wait_tensorcnt 0x2" ::: "memory");
}
__device__ __forceinline__ void wait_tensorcnt1() {
    asm volatile("s_wait_tensorcnt 0x1" ::: "memory");
}
__device__ __forceinline__ void wait_tensorcnt0() {
    asm volatile("s_wait_tensorcnt 0x0" ::: "memory");
}
__device__ __forceinline__ void wait_dscnt0() {
    asm volatile("s_wait_dscnt 0x0" ::: "memory");
}

#if HAVE_TDM
// Build a 2-D tensor descriptor (ISA cdna5 ch.8) and issue TENSOR_LOAD_TO_LDS.
// Tile: tile_w fp32 elements per row, tile_h rows, row stride = stride_elems.
// clang-23 toolchain builtin arity is 6:
//   (uint32x4 g0, int32x8 g1, int32x4 g2, int32x4 g3, int32x8 extra, i32 cpol)
// groups 0..3 follow the ISA D# layout; the extra operand is zero-filled
__device__ __forceinline__ void tdm_load_tile_2d(const void* gaddr, unsigned lds_addr,
                                                 unsigned tile_w, unsigned tile_h,
                                                 unsigned stride_elems)
{
    unsigned long long ga = (unsigned long long)(uintptr_t)gaddr;
    u32x4 g0;
    g0.x = 1u;                                    // count=1, user descriptor, no gather
    g0.y = lds_addr;                              // LDS byte address [63:32]
    g0.z = (unsigned)ga;                          // global addr [95:64]
    g0.w = (unsigned)((ga >> 32) & 0x01FFFFFFu)   // global addr [120:96]
         | (2u << 30);                            // type=2 ("image") [127:126]
    i32x8 g1;
    g1[0] = (int)(2u << 16);                                // data_size=2 -> 4 bytes
    g1[1] = (int)((D_ & 0xFFFFu) << 16);                    // abar_addr=0 | tensor_dim0 lo16
    g1[2] = (int)(((unsigned)D_ >> 16) | ((S_ & 0xFFFFu) << 16)); // dim0 hi16 | dim1 lo16
    g1[3] = (int)(((unsigned)S_ >> 16) | (tile_w << 16));   // dim1 hi16 | tile_dim0
    g1[4] = (int)tile_h;                                    // tile_dim1 (tile_dim2=0)
    g1[5] = (int)stride_elems;                              // tensor_dim0_stride lo32
    g1[6] = 0;                                              // stride0 hi16 | stride1 lo16
    g1[7] = 0;                                              // stride1 hi32
    i32x4 g2 = (i32x4){0, 0, 0, 0};                         // 2-D: groups 2/3 unused
    i32x4 g3 = (i32x4){0, 0, 0, 0};
    i32x8 gx = (i32x8){0, 0, 0, 0, 0, 0, 0, 0};             // extra operand: zero-filled
    __builtin_amdgcn_tensor_load_to_lds(g0, g1, g2, g3, gx, 0);
}
#endif

__device__ __forceinline__ float lif_step(float xt, float bm, float bs, float cad,
                                          float lth, float& mem, float& syn,
                                          float& ta, float& refr)
{
    syn = bs * syn + xt;
    float nr = (refr <= 0.0f) ? 1.0f : 0.0f;      // non-refractory gate
    mem = bm * mem + syn * nr;
    float th = lth + cad * ta;                     // cur_th = lth + p*scale*adapt
    float sp = (mem >= th) ? 1.0f : 0.0f;          // Heaviside spike
    mem = mem - sp * th;                           // reset: subtract
    ta  = 0.9f * ta + 0.1f * sp;                   // threshold-adapt EMA
    refr = (sp > 0.5f) ? 2.0f : fmaxf(refr - 1.0f, 0.0f);
    return sp;
}

__global__ __launch_bounds__(TPB) void lif_fused_kernel(
    const float* __restrict__ x,
    const float* __restrict__ p_beta_mem,
    const float* __restrict__ p_beta_syn,
    const float* __restrict__ p_adapt,
    const float* __restrict__ scale,
    const float* __restrict__ lth_in,
    float* __restrict__ out)
{
    const int b  = blockIdx.x / DCH;
    const int dc = blockIdx.x % DCH;
    const int d  = dc * TPB + (int)threadIdx.x;

    const float bm  = p_beta_mem[0];
    const float bs  = p_beta_syn[0];
    const float cad = p_adapt[0] * scale[d];
    const float lth = lth_in[d];

    float mem = 0.0f, syn = 0.0f, ta = 0.0f, refr = 0.0f;

    float* outp = out + (size_t)b * S_ * D_ + d;   // spikes: + t*D_ per step

#if HAVE_TDM
    // Per-wave private buffers: smem[buf][wave][tt][lane]
    __shared__ float smem[NBUF][NW][TT][32];
    // Low 32 bits of the generic address of an LDS object == LDS byte offset.
    const unsigned lds0 = (unsigned)(uintptr_t)&smem[0][0][0][0];
    // AS(3) view of the same buffer -> consumer loop emits ds loads.
    lds_f* sb = (lds_f*)lds0;

    const int wave = (int)threadIdx.x >> 5;
    const int lane = (int)threadIdx.x & 31;
    const float* gw = x + (size_t)b * S_ * D_ + dc * TPB + wave * 32;
    const unsigned lbufw  = (unsigned)(NW * TT * 32);      // floats per buffer
    const unsigned lwavew = (unsigned)(wave * TT * 32);    // float offset of this wave

    // Prime the pipeline: chunks 0..NBUF-1 in flight (TENSORcnt = NBUF).
#pragma unroll
    for (int c = 0; c < NBUF; ++c)
        tdm_load_tile_2d(gw + (size_t)c * TT * D_,
                         lds0 + (unsigned)c * lbufw * 4u + lwavew * 4u, 32u, TT, D_);

    for (int c = 0; c < NCH; ++c) {
        // TDM ops of one wave complete in order; issued = min(c+NBUF, NCH),
        // need completed >= c+1  =>  wait tensorcnt <= min(NBUF-1, NCH-1-c).
        if (c < NCH - 2)       wait_tensorcnt2();
        else if (c == NCH - 2) wait_tensorcnt1();
        else                   wait_tensorcnt0();

        const unsigned base = (unsigned)(c % NBUF) * lbufw + lwavew + (unsigned)lane;
        const size_t tbase = (size_t)c * TT;
#pragma unroll 8
        for (int tt = 0; tt < TT; ++tt) {
            float xt = sb[base + (unsigned)tt * 32u];
            float sp = lif_step(xt, bm, bs, cad, lth, mem, syn, ta, refr);
            __builtin_nontemporal_store(sp, &outp[(tbase + tt) * D_]);
        }
        if (c + NBUF < NCH) {
            // Our ds loads of this buffer must retire before TDM overwrites it.
            wait_dscnt0();
            tdm_load_tile_2d(gw + (size_t)(c + NBUF) * TT * D_,
                             lds0 + (unsigned)(c % NBUF) * lbufw * 4u + lwavew * 4u,
                             32u, TT, D_);
        }
    }
#else
    const float* xcol = x + (size_t)b * S_ * D_ + d;
    for (int t = 0; t < S_; ++t) {
        if (t + 16 < S_)
            __builtin_prefetch(xcol + (size_t)(t + 16) * D_, 0, 0);  // global_prefetch_b8
        float xt = xcol[(size_t)t * D_];
        float sp = lif_step(xt, bm, bs, cad, lth, mem, syn, ta, refr);
        __builtin_nontemporal_store(sp, &outp[(size_t)t * D_]);
    }
#endif

    // Final carries, concatenated after the spike train in return order.
    const size_t BD = (size_t)B_ * D_;
    const size_t c0 = (size_t)B_ * S_ * D_;
    const size_t bd = (size_t)b * D_ + d;
    out[c0 + bd]           = mem;
    out[c0 + BD + bd]      = syn;
    out[c0 + 2 * BD + bd]  = ta;
    out[c0 + 3 * BD + bd]  = refr;
}

extern "C" void kernel_launch(void* const* d_in, const int* in_sizes, int n_in,
                              void* d_out, int out_size, void* d_ws, size_t ws_size,
                              hipStream_t stream)
{
    const float* x   = (const float*)d_in[0];
    const float* bm  = (const float*)d_in[1];
    const float* bs  = (const float*)d_in[2];
    const float* ap  = (const float*)d_in[3];
    const float* sc  = (const float*)d_in[4];
    const float* lt  = (const float*)d_in[5];
    float* out = (float*)d_out;
    (void)in_sizes; (void)n_in; (void)out_size; (void)d_ws; (void)ws_size;

    dim3 grid(B_ * DCH);   // 128 blocks
    dim3 block(TPB);       // 128 threads = 4 wave32
    hipLaunchKernelGGL(lif_fused_kernel, grid, block, 0, stream,
                       x, bm, bs, ap, sc, lt, out);
}